// GNNPredictor_72756745994791
// MI455X (gfx1250) — compile-verified
//
#include <hip/hip_runtime.h>
#include <hip/hip_bf16.h>

// ---------------------------------------------------------------------------
// GNN predictor for MI455X (gfx1250, wave32).
// Dense GEMMs use V_WMMA_F32_16X16X4_F32 (fp32-native matrix pipe).
// W is staged in LDS as interleaved K-pairs so each B fragment is a single
// aligned ds_load_b64 feeding WMMA src1 directly.
// ---------------------------------------------------------------------------

typedef float v2f __attribute__((ext_vector_type(2)));
typedef float v8f __attribute__((ext_vector_type(8)));

#define N_NODES  100000
#define N_EDGES  640000
#define N_GRAPHS 4096
#define HID      128
#define BN_EPS   1e-5f

// ---------------------------------------------------------------------------
// Generic float zero
// ---------------------------------------------------------------------------
__global__ void zero_kernel(float* __restrict__ p, int n) {
    int i = blockIdx.x * blockDim.x + threadIdx.x;
    if (i < n) p[i] = 0.0f;
}

// ---------------------------------------------------------------------------
// Degree / normalization:  deg starts at 1 (self-loop), +1 per incoming edge,
// then dinv = rsqrt(deg).
// ---------------------------------------------------------------------------
__global__ void deg_init_kernel(float* __restrict__ deg, int n) {
    int i = blockIdx.x * blockDim.x + threadIdx.x;
    if (i < n) deg[i] = 1.0f;
}
__global__ void deg_accum_kernel(const int* __restrict__ dst, float* __restrict__ deg, int e) {
    int i = blockIdx.x * blockDim.x + threadIdx.x;
    if (i < e) atomicAdd(&deg[dst[i]], 1.0f);
}
__global__ void deg_rsqrt_kernel(float* __restrict__ deg, int n) {
    int i = blockIdx.x * blockDim.x + threadIdx.x;
    if (i < n) deg[i] = rsqrtf(deg[i]);
}

// ---------------------------------------------------------------------------
// WMMA GEMM:  Out[M x NCOLS] = A[M x 128] @ W[128 x NCOLS]   (all fp32)
// Block = 256 threads = 8 wave32.  Each block covers 128 rows; each wave owns
// a 16-row stripe and all NCOLS columns (NCOLS/16 accumulator tiles).
// K loop: 32 steps of V_WMMA_F32_16X16X4_F32.
//
// Fragment layouts (CDNA5 ISA 7.12.2):
//   A 16x4 f32 : v0 = {K=0 | K=2}, v1 = {K=1 | K=3} across lane halves, M=lane%16
//   B  4x16 f32: mirrored (K rows striped across vgprs/halves, N=lane%16)
//   C/D 16x16  : vgpr r -> M = r + 8*(lane>=16), N = lane%16
//
// LDS holds W as interleaved K-pairs: sWp[p][n] = {W[2p][n], W[2p+1][n]}.
// kb = k + 2*half is always even, so fragment = sWp[kb>>1][col] via one b64.
// ---------------------------------------------------------------------------
template <int NCOLS>
__global__ __launch_bounds__(256) void gemm_wmma_kernel(
        const float* __restrict__ A, const float* __restrict__ W,
        float* __restrict__ Out, int M) {
    constexpr int T = NCOLS / 16;
    __shared__ v2f sWp[64 * NCOLS];          // 64 K-pairs x NCOLS cols (64KB @128)

    const int tid  = threadIdx.x;
    const int wave = tid >> 5;
    const int lane = tid & 31;
    const int l16  = lane & 15;
    const int half = lane >> 4;              // 0: lanes 0-15, 1: lanes 16-31
    const int rowBase = blockIdx.x * 128;

    // Stage + interleave W: pair up rows 2p and 2p+1 per column.
    for (int i = tid; i < 64 * NCOLS; i += 256) {
        const int p = i / NCOLS;
        const int n = i % NCOLS;
        v2f v;
        v.x = W[(2 * p)     * NCOLS + n];
        v.y = W[(2 * p + 1) * NCOLS + n];
        sWp[p * NCOLS + n] = v;
    }
    __syncthreads();

    // A row for this lane (clamped so tail blocks never read OOB; stores are
    // masked instead — EXEC stays all-ones through every WMMA).
    int rowA = rowBase + wave * 16 + l16;
    if (rowA >= M) rowA = M - 1;
    const float* __restrict__ aRow = A + (size_t)rowA * 128;

    v8f acc[T];
#pragma unroll
    for (int t = 0; t < T; ++t) acc[t] = (v8f){0.f,0.f,0.f,0.f,0.f,0.f,0.f,0.f};

    for (int k = 0; k < 128; k += 4) {
        const int kb = k + 2 * half;         // even
        const int p  = kb >> 1;
        v2f a;
        a.x = aRow[kb];
        a.y = aRow[kb + 1];
        const v2f* __restrict__ bRow = sWp + p * NCOLS;
#pragma unroll
        for (int t = 0; t < T; ++t) {
            const v2f b = bRow[t * 16 + l16];   // single ds_load_b64
            acc[t] = __builtin_amdgcn_wmma_f32_16x16x4_f32(
                false, a, false, b, (short)0, acc[t], false, false);
        }
    }

    // Store D: vgpr r holds row (r + 8*half) of the 16-row stripe, col = lane%16.
    const int row0 = rowBase + wave * 16 + 8 * half;
#pragma unroll
    for (int t = 0; t < T; ++t) {
        const int col = t * 16 + l16;
#pragma unroll
        for (int r = 0; r < 8; ++r) {
            const int row = row0 + r;
            if (row < M) Out[(size_t)row * NCOLS + col] = acc[t][r];
        }
    }
}

// ---------------------------------------------------------------------------
// Edge aggregation:  out[dst] += hw[src] * dinv[src]*dinv[dst]
// One wave per edge (edges = E real edges + N self-loops); each lane moves a
// float4 of channels and issues 4 global_atomic_add_f32.
// ---------------------------------------------------------------------------
__global__ void aggregate_kernel(const float* __restrict__ hw,
                                 const int* __restrict__ eidx,   // [2, E]
                                 const float* __restrict__ dinv,
                                 float* __restrict__ out,
                                 int nEdges, int nNodes) {
    const int t = blockIdx.x * blockDim.x + threadIdx.x;
    const int e = t >> 5;
    if (e >= nEdges + nNodes) return;
    const int c = (t & 31) << 2;

    int s, d;
    if (e < nEdges) { s = eidx[e]; d = eidx[nEdges + e]; }
    else            { s = d = e - nEdges; }                 // self-loop

    const float nm = dinv[s] * dinv[d];
    const float4 v = *(const float4*)(hw + (size_t)s * HID + c);
    float* o = out + (size_t)d * HID + c;
    atomicAdd(o + 0, v.x * nm);
    atomicAdd(o + 1, v.y * nm);
    atomicAdd(o + 2, v.z * nm);
    atomicAdd(o + 3, v.w * nm);
}

// ---------------------------------------------------------------------------
// BatchNorm pass 1: per-channel sum & sum-of-squares into stats[0..ch-1] and
// stats[ch..2ch-1] via atomics (each thread owns one channel over a row slab).
// ---------------------------------------------------------------------------
__global__ void bn_stats_kernel(const float* __restrict__ h, float* __restrict__ stats,
                                int rows, int ch) {
    const int c   = threadIdx.x % ch;
    const int rpb = blockDim.x / ch;
    int r = blockIdx.x * rpb + threadIdx.x / ch;
    const int stride = gridDim.x * rpb;
    float s = 0.f, s2 = 0.f;
    for (; r < rows; r += stride) {
        const float v = h[(size_t)r * ch + c];
        s += v; s2 += v * v;
    }
    atomicAdd(&stats[c],      s);
    atomicAdd(&stats[ch + c], s2);
}

// ---------------------------------------------------------------------------
// BatchNorm pass 2 + ReLU (in place).  Biased variance = E[x^2] - mu^2,
// matching the reference.  (Pre-BN biases cancel and are dropped.)
// ---------------------------------------------------------------------------
__global__ void bn_apply_relu_kernel(float* __restrict__ h, const float* __restrict__ stats,
                                     const float* __restrict__ gamma,
                                     const float* __restrict__ beta,
                                     int rows, int ch) {
    const int idx = blockIdx.x * blockDim.x + threadIdx.x;
    if (idx >= rows * ch) return;
    const int c = idx % ch;
    const float inv = 1.0f / (float)rows;
    const float mu  = stats[c] * inv;
    const float var = stats[ch + c] * inv - mu * mu;
    const float sc  = gamma[c] * rsqrtf(var + BN_EPS);
    const float v   = (h[idx] - mu) * sc + beta[c];
    h[idx] = v > 0.f ? v : 0.f;
}

// ---------------------------------------------------------------------------
// Global mean pool: atomic accumulate per graph, then divide by counts.
// ---------------------------------------------------------------------------
__global__ void pool_accum_kernel(const float* __restrict__ h, const int* __restrict__ batch,
                                  float* __restrict__ pooled, float* __restrict__ counts,
                                  int n) {
    const int t = blockIdx.x * blockDim.x + threadIdx.x;
    const int node = t >> 5;
    if (node >= n) return;
    const int c = (t & 31) << 2;
    const int g = batch[node];
    const float4 v = *(const float4*)(h + (size_t)node * HID + c);
    float* o = pooled + (size_t)g * HID + c;
    atomicAdd(o + 0, v.x);
    atomicAdd(o + 1, v.y);
    atomicAdd(o + 2, v.z);
    atomicAdd(o + 3, v.w);
    if ((t & 31) == 0) atomicAdd(&counts[g], 1.0f);
}
__global__ void pool_div_kernel(float* __restrict__ pooled, const float* __restrict__ counts) {
    const int idx = blockIdx.x * blockDim.x + threadIdx.x;
    if (idx >= N_GRAPHS * HID) return;
    pooled[idx] /= fmaxf(counts[idx >> 7], 1.0f);
}

// ---------------------------------------------------------------------------
// Final head:  out[g] = dot(hid[g, :64], Wm1) + bm1
// ---------------------------------------------------------------------------
__global__ void head_final_kernel(const float* __restrict__ hid, const float* __restrict__ Wm1,
                                  const float* __restrict__ bm1, float* __restrict__ out) {
    const int g = blockIdx.x * blockDim.x + threadIdx.x;
    if (g >= N_GRAPHS) return;
    float s = bm1[0];
#pragma unroll
    for (int c = 0; c < 64; ++c) s += hid[g * 64 + c] * Wm1[c];
    out[g] = s;
}

// ---------------------------------------------------------------------------
// Host-side orchestration
// ---------------------------------------------------------------------------
extern "C" void kernel_launch(void* const* d_in, const int* in_sizes, int n_in,
                              void* d_out, int out_size, void* d_ws, size_t ws_size,
                              hipStream_t stream) {
    (void)in_sizes; (void)n_in; (void)out_size; (void)ws_size;

    const float* x      = (const float*)d_in[0];   // [N, 128]
    const int*   eidx   = (const int*)  d_in[1];   // [2, E]
    const int*   batch  = (const int*)  d_in[2];   // [N]
    const float* Wg     = (const float*)d_in[3];   // [3, 128, 128]
    /* bg = d_in[4]      : cancels inside training-mode BN */
    const float* gamma  = (const float*)d_in[5];   // [3, 128]
    const float* beta   = (const float*)d_in[6];   // [3, 128]
    const float* Wm0    = (const float*)d_in[7];   // [128, 64]
    /* bm0 = d_in[8]     : cancels inside training-mode BN */
    const float* gm0    = (const float*)d_in[9];   // [64]
    const float* betam0 = (const float*)d_in[10];  // [64]
    const float* Wm1    = (const float*)d_in[11];  // [64, 1]
    const float* bm1    = (const float*)d_in[12];  // [1]
    float* out = (float*)d_out;                    // [4096, 1]

    // Workspace layout (all float, 16B-aligned chunks): ~106 MB total.
    float* wsf    = (float*)d_ws;
    float* dinv   = wsf;                                   // N_NODES
    float* hA     = dinv   + N_NODES;                      // N_NODES * 128
    float* hB     = hA     + (size_t)N_NODES * HID;        // N_NODES * 128
    float* stats  = hB     + (size_t)N_NODES * HID;        // 256
    float* pooled = stats  + 256;                          // N_GRAPHS * 128
    float* counts = pooled + (size_t)N_GRAPHS * HID;       // N_GRAPHS
    float* hid    = counts + N_GRAPHS;                     // N_GRAPHS * 64

    const int NC     = HID * N_NODES;                      // 12.8M
    const int EDGES  = N_EDGES + N_NODES;                  // incl. self-loops

    // --- symmetric GCN normalization -------------------------------------
    deg_init_kernel <<<(N_NODES + 255) / 256, 256, 0, stream>>>(dinv, N_NODES);
    deg_accum_kernel<<<(N_EDGES + 255) / 256, 256, 0, stream>>>(eidx + N_EDGES, dinv, N_EDGES);
    deg_rsqrt_kernel<<<(N_NODES + 255) / 256, 256, 0, stream>>>(dinv, N_NODES);

    // --- 3 GCN layers ----------------------------------------------------
    const float* in = x;
    for (int i = 0; i < 3; ++i) {
        gemm_wmma_kernel<HID><<<(N_NODES + 127) / 128, 256, 0, stream>>>(
            in, Wg + (size_t)i * HID * HID, hB, N_NODES);
        zero_kernel<<<(NC + 255) / 256, 256, 0, stream>>>(hA, NC);
        aggregate_kernel<<<(EDGES * 32 + 255) / 256, 256, 0, stream>>>(
            hB, eidx, dinv, hA, N_EDGES, N_NODES);
        zero_kernel<<<1, 256, 0, stream>>>(stats, 256);
        bn_stats_kernel<<<256, 256, 0, stream>>>(hA, stats, N_NODES, HID);
        bn_apply_relu_kernel<<<(NC + 255) / 256, 256, 0, stream>>>(
            hA, stats, gamma + i * HID, beta + i * HID, N_NODES, HID);
        in = hA;
    }

    // --- global mean pool ------------------------------------------------
    zero_kernel<<<(N_GRAPHS * HID + 255) / 256, 256, 0, stream>>>(pooled, N_GRAPHS * HID);
    zero_kernel<<<(N_GRAPHS + 255) / 256, 256, 0, stream>>>(counts, N_GRAPHS);
    pool_accum_kernel<<<(N_NODES * 32 + 255) / 256, 256, 0, stream>>>(
        hA, batch, pooled, counts, N_NODES);
    pool_div_kernel<<<(N_GRAPHS * HID + 255) / 256, 256, 0, stream>>>(pooled, counts);

    // --- MLP head --------------------------------------------------------
    gemm_wmma_kernel<64><<<(N_GRAPHS + 127) / 128, 256, 0, stream>>>(
        pooled, Wm0, hid, N_GRAPHS);
    zero_kernel<<<1, 256, 0, stream>>>(stats, 256);
    bn_stats_kernel<<<64, 256, 0, stream>>>(hid, stats, N_GRAPHS, 64);
    bn_apply_relu_kernel<<<(N_GRAPHS * 64 + 255) / 256, 256, 0, stream>>>(
        hid, stats, gm0, betam0, N_GRAPHS, 64);
    head_final_kernel<<<(N_GRAPHS + 255) / 256, 256, 0, stream>>>(hid, Wm1, bm1, out);
}